// ProportionalMasking_cumsum_9646496547388
// MI455X (gfx1250) — compile-verified
//
#include <hip/hip_runtime.h>
#include <hip/hip_bf16.h>
#include <cstdint>

// Problem constants (from reference: B,N,D = 16,2048,1024; TEMPERATURE=2.0)
constexpr int   Bc   = 16;
constexpr int   Nc   = 2048;
constexpr int   Dc   = 1024;
constexpr float TEMP = 2.0f;

constexpr int NSEG = 64;            // segments along N
constexpr int SEGN = 32;            // rows per segment  (NSEG*SEGN == Nc)
constexpr int TPB  = 256;           // 8 wave32 per block
constexpr int DBLK = Dc / TPB;      // 4 column-blocks per (b,segment)

typedef float v4f __attribute__((ext_vector_type(4)));

// ---------------------------------------------------------------------------
// Wave-local async staging: copy a SEGN x 32-column f32 tile (row stride = Dc
// floats in global) into this wave's LDS region using CDNA5 async loads.
// Each lane moves 16B per instruction: 32 lanes cover 4 rows x 128B per issue,
// 8 issues cover the 32-row tile.  ASYNCcnt tracks completion (wave-local, so
// no workgroup barrier is needed before the same wave consumes the tile).
// ---------------------------------------------------------------------------
__device__ __forceinline__ void stage_tile_async(const float* gbase,
                                                 uint32_t lds_byte_base,
                                                 int lane) {
    const int rsub = lane >> 3;          // 0..3  (row within group of 4)
    const int c4   = (lane & 7) << 2;    // 0,4,...,28 (float column of 16B chunk)
#pragma unroll
    for (int r = 0; r < SEGN / 4; ++r) {
        const int row = (r << 2) + rsub;
        const uint64_t gaddr =
            (uint64_t)(uintptr_t)(gbase + (size_t)row * Dc + c4);
        // LDS byte offset of tile[row][c4] (wave-relative; low 32 bits of a
        // shared-aperture generic pointer are the LDS offset per the ISA).
        const uint32_t laddr =
            lds_byte_base + (uint32_t)((((row << 5) + c4) << 2));
        asm volatile("global_load_async_to_lds_b128 %0, %1, off"
                     :: "v"(laddr), "v"(gaddr)
                     : "memory");
    }
    asm volatile("s_wait_asynccnt 0x0" ::: "memory");
}

__device__ __forceinline__ void decode_seg(int blk, int& b, int& s, int& dblk) {
    dblk = blk % DBLK;                  // powers of two -> shifts/ands
    s    = (blk / DBLK) % NSEG;
    b    = blk / (DBLK * NSEG);
}

// K1: per-segment sum of |x|  ->  P[(b*NSEG+s)*Dc + d]
__global__ void k_seg_abs_sum(const float* __restrict__ x,
                              float* __restrict__ P) {
    __shared__ float tile[TPB / 32][SEGN][32];
    int b, s, dblk;
    decode_seg(blockIdx.x, b, s, dblk);
    const int w    = threadIdx.x >> 5;
    const int lane = threadIdx.x & 31;
    const int d0w  = dblk * TPB + w * 32;
    const float* gbase = x + ((size_t)b * Nc + (size_t)s * SEGN) * Dc + d0w;
    const uint32_t ldsbase = (uint32_t)(uintptr_t)(&tile[w][0][0]);

    stage_tile_async(gbase, ldsbase, lane);

    float acc = 0.f;
#pragma unroll
    for (int n = 0; n < SEGN; ++n) acc += fabsf(tile[w][n][lane]);

    const int d = d0w + lane;
    P[((size_t)b * NSEG + s) * Dc + d] = acc;
}

// K2: column totals across segments: S1[b*Dc+d] = sum_s P[..]
__global__ void k_colsum(const float* __restrict__ P, float* __restrict__ S) {
    const int i = blockIdx.x * blockDim.x + threadIdx.x;   // 0 .. Bc*Dc-1
    const int b = i / Dc, d = i % Dc;
    float acc = 0.f;
#pragma unroll 8
    for (int q = 0; q < NSEG; ++q)
        acc += P[((size_t)b * NSEG + q) * Dc + d];
    S[i] = acc;
}

// K3: per-segment sum of exp(|x| * T / S1)  ->  P (overwrites abs-sums)
__global__ void k_seg_exp_sum(const float* __restrict__ x,
                              const float* __restrict__ S1,
                              float* __restrict__ P) {
    __shared__ float tile[TPB / 32][SEGN][32];
    int b, s, dblk;
    decode_seg(blockIdx.x, b, s, dblk);
    const int w    = threadIdx.x >> 5;
    const int lane = threadIdx.x & 31;
    const int d0w  = dblk * TPB + w * 32;
    const int d    = d0w + lane;
    const float* gbase = x + ((size_t)b * Nc + (size_t)s * SEGN) * Dc + d0w;
    const uint32_t ldsbase = (uint32_t)(uintptr_t)(&tile[w][0][0]);

    const float scale = TEMP / S1[b * Dc + d];

    stage_tile_async(gbase, ldsbase, lane);

    float acc = 0.f;
#pragma unroll
    for (int n = 0; n < SEGN; ++n)
        acc += __expf(fabsf(tile[w][n][lane]) * scale);

    P[((size_t)b * NSEG + s) * Dc + d] = acc;
}

// K4: per-segment count of prefix-sums <= (rv - 0.001)*S2, seeded with the
// exclusive prefix of the segment exp-sums.  The single walk over all NSEG
// segment sums yields BOTH the exclusive prefix (q < s) and the column total
// S2 (all q), so no separate S2 reduction kernel/buffer is needed.
// Monotone cumsum => global count is the sum of per-segment counts.
__global__ void k_seg_count(const float* __restrict__ x,
                            const float* __restrict__ P,
                            const float* __restrict__ S1,
                            const float* __restrict__ rv,
                            unsigned* __restrict__ CNT) {
    __shared__ float tile[TPB / 32][SEGN][32];
    int b, s, dblk;
    decode_seg(blockIdx.x, b, s, dblk);
    const int w    = threadIdx.x >> 5;
    const int lane = threadIdx.x & 31;
    const int d0w  = dblk * TPB + w * 32;
    const int d    = d0w + lane;
    const int i    = b * Dc + d;
    const float* gbase = x + ((size_t)b * Nc + (size_t)s * SEGN) * Dc + d0w;
    const uint32_t ldsbase = (uint32_t)(uintptr_t)(&tile[w][0][0]);

    // One walk over the (L2-resident) segment exp-sums: exclusive prefix + S2.
    float run = 0.f, tot = 0.f;
    for (int q = 0; q < NSEG; ++q) {
        const float v = P[((size_t)b * NSEG + q) * Dc + d];
        tot += v;
        if (q < s) run += v;
    }

    const float thresh = (rv[i] - 0.001f) * tot;  // cum<=rv  <=>  prefix<=thresh
    const float scale  = TEMP / S1[i];

    stage_tile_async(gbase, ldsbase, lane);

    unsigned cnt = 0;
#pragma unroll
    for (int n = 0; n < SEGN; ++n) {
        run += __expf(fabsf(tile[w][n][lane]) * scale);
        cnt += (run <= thresh) ? 1u : 0u;
    }
    atomicAdd(&CNT[i], cnt);
}

// K5: gathered = |x[b, clamp(cnt, 0, N-1), d]|
__global__ void k_gather(const float* __restrict__ x,
                         const unsigned* __restrict__ CNT,
                         float* __restrict__ G) {
    const int i = blockIdx.x * blockDim.x + threadIdx.x;
    const int b = i / Dc, d = i % Dc;
    unsigned idx = CNT[i];
    if (idx > (unsigned)(Nc - 1)) idx = Nc - 1;       // JAX gather clamps OOB
    G[i] = fabsf(x[((size_t)b * Nc + idx) * Dc + d]);
}

// K6: out = x * (|x| >= gathered), float4-vectorized streaming pass.
// Last use of x and write-once out: use non-temporal hints (TH=NT) so this
// 256 MiB streaming pass does not evict the reused data (x copies already
// served; G stays hot) from the 192 MB L2.
__global__ void k_mask_write(const v4f* __restrict__ x4,
                             const float* __restrict__ G,
                             v4f* __restrict__ o4) {
    const size_t t = (size_t)blockIdx.x * blockDim.x + threadIdx.x;
    const size_t e = t << 2;                           // element index
    const int d = (int)(e % Dc);
    const int b = (int)(e / ((size_t)Nc * Dc));
    const v4f xv = __builtin_nontemporal_load(x4 + t);
    const v4f gv = *(const v4f*)(G + (size_t)b * Dc + d);
    v4f ov;
    ov.x = (fabsf(xv.x) >= gv.x) ? xv.x : 0.f;
    ov.y = (fabsf(xv.y) >= gv.y) ? xv.y : 0.f;
    ov.z = (fabsf(xv.z) >= gv.z) ? xv.z : 0.f;
    ov.w = (fabsf(xv.w) >= gv.w) ? xv.w : 0.f;
    __builtin_nontemporal_store(ov, o4 + t);
}

__global__ void k_zero_u32(unsigned* __restrict__ p, int n) {
    const int i = blockIdx.x * blockDim.x + threadIdx.x;
    if (i < n) p[i] = 0u;
}

extern "C" void kernel_launch(void* const* d_in, const int* in_sizes, int n_in,
                              void* d_out, int out_size, void* d_ws, size_t ws_size,
                              hipStream_t stream) {
    const float* x  = (const float*)d_in[0];   // (B,N,D) f32
    const float* rv = (const float*)d_in[1];   // (B,D)   f32

    // Workspace layout (floats): S1 | G | CNT(u32) | P[NSEG*Bc*Dc]
    float*    ws  = (float*)d_ws;
    float*    S1  = ws;
    float*    G   = ws + (size_t)Bc * Dc;
    unsigned* CNT = (unsigned*)(ws + (size_t)2 * Bc * Dc);
    float*    P   = ws + (size_t)3 * Bc * Dc;

    const int cols = Bc * Dc;                      // 16384 columns
    const dim3 segGrid(Bc * NSEG * DBLK);          // 4096 blocks
    const dim3 colGrid(cols / 256);                // 64 blocks

    k_zero_u32   <<<colGrid, 256, 0, stream>>>(CNT, cols);
    k_seg_abs_sum<<<segGrid, TPB, 0, stream>>>(x, P);
    k_colsum     <<<colGrid, 256, 0, stream>>>(P, S1);
    k_seg_exp_sum<<<segGrid, TPB, 0, stream>>>(x, S1, P);
    k_seg_count  <<<segGrid, TPB, 0, stream>>>(x, P, S1, rv, CNT);
    k_gather     <<<colGrid, 256, 0, stream>>>(x, CNT, G);

    const size_t quads = (size_t)Bc * Nc * Dc / 4;  // 8,388,608
    k_mask_write <<<(unsigned)(quads / 256), 256, 0, stream>>>(
        (const v4f*)x, G, (v4f*)d_out);
}